// LinearSelfAttention_29841432772830
// MI455X (gfx1250) — compile-verified
//
#include <hip/hip_runtime.h>

// fp32 linear attention via reassociation: Out = Q @ (K^T @ V), per head.
// B=1, S=4096, H=16, D=64.  qkv layout: [S][3][H][D] -> idx = s*3072 + t*1024 + h*64 + d

typedef __attribute__((ext_vector_type(2))) float v2f;
typedef __attribute__((ext_vector_type(8))) float v8f;

#define SEQ      4096
#define NH       16
#define HD       64
#define ROWSTR   3072          // floats per sequence position (3*16*64)
#define NCHUNK   32
#define CH       (SEQ / NCHUNK)  // 128 rows per chunk

// ---------------------------------------------------------------------------
// Kernel 1: per-(chunk, head) partial KV = K_chunk^T @ V_chunk  (64x64 f32)
// 16 waves per block; wave (i,j) owns KV tile rows 16i..16i+15, cols 16j..16j+15.
// A-frag (16x4 f32): lane r,half -> A[r][2*half], A[r][2*half+1]
// B-frag (4x16 f32): lane r,half -> B[2*half][r], B[2*half+1][r]
// ---------------------------------------------------------------------------
__global__ __launch_bounds__(512)
void kv_partial_kernel(const float* __restrict__ qkv, float* __restrict__ part) {
    const int chunk = blockIdx.x;
    const int h     = blockIdx.y;
    const int w     = threadIdx.x >> 5;
    const int lane  = threadIdx.x & 31;
    const int hi    = lane >> 4;       // half-wave
    const int r     = lane & 15;
    const int d1b   = (w >> 2) << 4;   // KV row tile
    const int d2b   = (w & 3)  << 4;   // KV col tile

    // K element: qkv[s*3072 + 1024 + h*64 + d1],  V element: +2048
    const float* kp = qkv + (size_t)h * HD + 1024 + d1b + r;
    const float* vp = qkv + (size_t)h * HD + 2048 + d2b + r;

    v8f acc = {};
    const int s0 = chunk * CH;
#pragma unroll 4
    for (int s = s0; s < s0 + CH; s += 4) {
        const size_t row = (size_t)(s + 2 * hi) * ROWSTR;
        v2f a, b;
        a.x = kp[row];
        a.y = kp[row + ROWSTR];
        b.x = vp[row];
        b.y = vp[row + ROWSTR];
        acc = __builtin_amdgcn_wmma_f32_16x16x4_f32(
            /*neg_a=*/false, a, /*neg_b=*/false, b,
            /*c_mod=*/(short)0, acc, /*reuse_a=*/false, /*reuse_b=*/false);
    }

    // C layout: vgpr rr -> M = d1b + rr + 8*hi, N = d2b + r
    float* o = part + ((size_t)(chunk * NH + h)) * 4096 + d2b + r;
#pragma unroll
    for (int rr = 0; rr < 8; ++rr) {
        const int d1 = d1b + rr + 8 * hi;
        o[d1 * 64] = acc[rr];
    }
}

// ---------------------------------------------------------------------------
// Kernel 2: deterministic fixed-order reduction of NCHUNK partials per head.
// 65536 elements total (16 heads * 64*64).
// ---------------------------------------------------------------------------
__global__ __launch_bounds__(256)
void kv_reduce_kernel(const float* __restrict__ part, float* __restrict__ kv) {
    const int i = blockIdx.x * 256 + threadIdx.x;   // h*4096 + e
    float s = 0.0f;
#pragma unroll
    for (int c = 0; c < NCHUNK; ++c)
        s += part[(size_t)c * NH * 4096 + i];
    kv[i] = s;
}

// ---------------------------------------------------------------------------
// Kernel 3: Out_block = Q_block(64x64) @ KV(64x64).  KV staged in LDS.
// 16 waves per block: wave (mi,j) -> 16 Q rows x 16 output cols, 16 WMMAs over K=64.
// ---------------------------------------------------------------------------
__global__ __launch_bounds__(512)
void out_kernel(const float* __restrict__ qkv, const float* __restrict__ kv,
                float* __restrict__ out) {
    __shared__ float lkv[64 * 64];

    const int sblk = blockIdx.x;   // 0..63 (64 Q rows each)
    const int h    = blockIdx.y;

    const float* kvh = kv + (size_t)h * 4096;
    for (int i = threadIdx.x; i < 4096; i += 512) lkv[i] = kvh[i];
    __syncthreads();

    const int w    = threadIdx.x >> 5;
    const int lane = threadIdx.x & 31;
    const int hi   = lane >> 4;
    const int r    = lane & 15;
    const int mi   = w >> 2;
    const int j    = w & 3;
    const int rowbase = sblk * 64 + mi * 16;
    const int d2b     = j * 16;

    // Q element: qkv[s*3072 + 0 + h*64 + d]; this lane reads row (rowbase + r)
    const float* qrow = qkv + (size_t)(rowbase + r) * ROWSTR + (size_t)h * HD;

    v8f acc = {};
#pragma unroll
    for (int k0 = 0; k0 < 64; k0 += 4) {
        v2f a, b;
        a = *(const v2f*)(qrow + k0 + 2 * hi);            // 8B-aligned pair
        b.x = lkv[(k0 + 2 * hi)     * 64 + d2b + r];
        b.y = lkv[(k0 + 2 * hi + 1) * 64 + d2b + r];
        acc = __builtin_amdgcn_wmma_f32_16x16x4_f32(
            false, a, false, b, (short)0, acc, false, false);
    }

    // Out[s][h][d] with s = rowbase + rr + 8*hi, d = d2b + r
#pragma unroll
    for (int rr = 0; rr < 8; ++rr) {
        const int grow = rowbase + rr + 8 * hi;
        out[((size_t)grow * NH + h) * HD + d2b + r] = acc[rr];
    }
}

// ---------------------------------------------------------------------------
extern "C" void kernel_launch(void* const* d_in, const int* in_sizes, int n_in,
                              void* d_out, int out_size, void* d_ws, size_t ws_size,
                              hipStream_t stream) {
    (void)in_sizes; (void)n_in; (void)out_size; (void)ws_size;
    const float* qkv = (const float*)d_in[0];
    float* out  = (float*)d_out;
    float* part = (float*)d_ws;                                  // NCHUNK*16*4096 floats
    float* kvf  = part + (size_t)NCHUNK * NH * 4096;             // 16*4096 floats

    dim3 g1(NCHUNK, NH);
    kv_partial_kernel<<<g1, 512, 0, stream>>>(qkv, part);

    kv_reduce_kernel<<<(NH * 4096) / 256, 256, 0, stream>>>(part, kvf);

    dim3 g3(SEQ / 64, NH);
    out_kernel<<<g3, 512, 0, stream>>>(qkv, kvf, out);
}